// HaversineSmoothedLoss_62010737820263
// MI455X (gfx1250) — compile-verified
//
#include <hip/hip_runtime.h>
#include <math.h>

// ---------------------------------------------------------------------------
// HaversineSmoothedLoss, fully fused for MI455X (gfx1250, wave32).
//
// dist[b,c] = 2R asin( sqrt( (1 - u_b . v_c) / 2 ) )  with u,v unit vectors
//   => the B x C distance matrix is an f32 GEMM with K=3
//   => V_WMMA_F32_16X16X4_F32 (K padded to 4 with zeros).
// dist_n cancels by softmax shift invariance; we do online max-rescaled
// accumulation instead (required: exp(-dist/tau) spans ~e^-267).
// loss_b = LSE(logits_b) - (sum e^{s-m} l)/(sum e^{s-m}),  s = -dist/tau
//   => ONE streaming pass over the 164MB logits, no B x C intermediate.
// ---------------------------------------------------------------------------

typedef __attribute__((ext_vector_type(2))) float v2f;
typedef __attribute__((ext_vector_type(8))) float v8f;

#define EARTH_R   6371.0f
#define TAU_SM    75.0f
#define DEG2RAD   0.017453292519943295f

// Precompute unit sphere vectors (x,y,z,0) from (lat,lon) degrees.
__global__ void prep_unitvec(const float* __restrict__ latlon, int n,
                             float* __restrict__ out) {
    int i = blockIdx.x * blockDim.x + threadIdx.x;
    if (i >= n) return;
    float lat = latlon[2 * i + 0] * DEG2RAD;
    float lon = latlon[2 * i + 1] * DEG2RAD;
    float cl = cosf(lat), sl = sinf(lat);
    float co = cosf(lon), so = sinf(lon);
    out[4 * i + 0] = cl * co;
    out[4 * i + 1] = cl * so;
    out[4 * i + 2] = sl;
    out[4 * i + 3] = 0.0f;   // K=3 pad for the 16x16x4 WMMA
}

// One workgroup = 4 wave32 = one 16-row block. Each wave strides column tiles,
// 2 tiles per iteration => two independent v_wmma in flight (hides the
// WMMA->VALU RAW latency, ISA 7.12.1) plus better load/EXP overlap.
__global__ __launch_bounds__(128)
void hav_loss_main(const float* __restrict__ pred,
                   const float* __restrict__ Uv,   // [B][4]
                   const float* __restrict__ Vv,   // [C][4]
                   float* __restrict__ row_loss,   // [B]
                   int C) {
    const int row0 = blockIdx.x * 16;
    const int lane = threadIdx.x & 31;
    const int wave = threadIdx.x >> 5;
    const int lo   = lane & 15;
    const int hi   = lane >> 4;      // 0: K=0,1 half ; 1: K=2,3 half

    // A operand, 16x4 f32 (ISA 7.12.2): lanes 0-15 hold (K0,K1) of row M=lane,
    // lanes 16-31 hold (K2,K3). Loaded once per workgroup.
    v2f avec;
    avec.x = Uv[(row0 + lo) * 4 + hi * 2 + 0];
    avec.y = Uv[(row0 + lo) * 4 + hi * 2 + 1];

    // Online accumulators: per lane, 8 rows (D-matrix VGPR j -> row j + 8*hi).
    float m[8], Z[8], W[8], ml[8], Zl[8];
#pragma unroll
    for (int j = 0; j < 8; ++j) {
        m[j] = -INFINITY; Z[j] = 0.0f; W[j] = 0.0f;
        ml[j] = -INFINITY; Zl[j] = 0.0f;
    }

    const long rowbase = (long)(row0 + 8 * hi) * C;   // first row this lane reads

    auto process_tile = [&](int t) {
        const int c0  = t * 16;
        const int col = c0 + lo;

        // B operand, 4x16 f32: VGPR0 = K0 (lanes 0-15) / K2 (16-31),
        // VGPR1 = K1 / K3.
        v2f bvec;
        bvec.x = Vv[col * 4 + hi * 2 + 0];
        bvec.y = Vv[col * 4 + hi * 2 + 1];

        // Prefetch logits a few tiles ahead (global_prefetch_b8).
        __builtin_prefetch(&pred[rowbase + c0 + 128], 0, 1);

        v8f acc = {};
        // 16x16 tile of u_b . v_c  (K=4, 4th component zero-padded).
        acc = __builtin_amdgcn_wmma_f32_16x16x4_f32(
            /*neg_a=*/false, avec, /*neg_b=*/false, bvec,
            /*c_mod=*/(short)0, acc, /*reuse_a=*/false, /*reuse_b=*/false);

#pragma unroll
        for (int j = 0; j < 8; ++j) {
            const int r = row0 + j + 8 * hi;
            float lv  = pred[(long)r * C + col];

            float dot = acc[j];
            float ah  = 0.5f * (1.0f - dot);           // haversine "a"
            ah = fminf(fmaxf(ah, 0.0f), 1.0f);
            float d   = (2.0f * EARTH_R) * asinf(__builtin_amdgcn_sqrtf(ah));
            float s   = -d * (1.0f / TAU_SM);

            // target softmax numerator/denominator, online max rescale
            float mm = fmaxf(m[j], s);
            float sc = __expf(m[j] - mm);              // native v_exp_f32
            float e  = __expf(s - mm);
            Z[j] = Z[j] * sc + e;
            W[j] = W[j] * sc + e * lv;
            m[j] = mm;

            // logits LSE, online max rescale
            float mm2 = fmaxf(ml[j], lv);
            float sc2 = __expf(ml[j] - mm2);
            float e2  = __expf(lv - mm2);
            Zl[j] = Zl[j] * sc2 + e2;
            ml[j] = mm2;
        }
    };

    const int nt = C >> 4;                 // 625 column tiles
    int t = wave;
    for (; t + 4 < nt; t += 8) {           // 2 tiles/iter, uniform per wave
        process_tile(t);
        process_tile(t + 4);
    }
    if (t < nt) process_tile(t);           // odd-count tail (no lane divergence)

    // Reduce across the 16 lanes sharing each row (xor masks stay in halves).
#pragma unroll
    for (int j = 0; j < 8; ++j) {
#pragma unroll
        for (int mask = 8; mask >= 1; mask >>= 1) {
            float om = __shfl_xor(m[j], mask, 32);
            float oZ = __shfl_xor(Z[j], mask, 32);
            float oW = __shfl_xor(W[j], mask, 32);
            float mm = fmaxf(m[j], om);
            float a  = __expf(m[j] - mm), b = __expf(om - mm);
            Z[j] = Z[j] * a + oZ * b;
            W[j] = W[j] * a + oW * b;
            m[j] = mm;

            float oml = __shfl_xor(ml[j], mask, 32);
            float oZl = __shfl_xor(Zl[j], mask, 32);
            float mm2 = fmaxf(ml[j], oml);
            Zl[j] = Zl[j] * __expf(ml[j] - mm2) + oZl * __expf(oml - mm2);
            ml[j] = mm2;
        }
    }

    // Cross-wave combine via LDS (4 waves x 16 rows x 5 scalars).
    __shared__ float red[4][16][5];
    if (lo == 0) {   // lanes 0 and 16 of each wave
#pragma unroll
        for (int j = 0; j < 8; ++j) {
            int r = j + 8 * hi;
            red[wave][r][0] = m[j];  red[wave][r][1] = Z[j];
            red[wave][r][2] = W[j];  red[wave][r][3] = ml[j];
            red[wave][r][4] = Zl[j];
        }
    }
    __syncthreads();

    if (threadIdx.x < 16) {
        int r = threadIdx.x;
        float fm = red[0][r][0], fZ = red[0][r][1], fW = red[0][r][2];
        float fml = red[0][r][3], fZl = red[0][r][4];
        for (int w = 1; w < 4; ++w) {
            float om = red[w][r][0], oZ = red[w][r][1], oW = red[w][r][2];
            float mm = fmaxf(fm, om);
            float a = __expf(fm - mm), b = __expf(om - mm);
            fZ = fZ * a + oZ * b;
            fW = fW * a + oW * b;
            fm = mm;
            float oml = red[w][r][3], oZl = red[w][r][4];
            float mm2 = fmaxf(fml, oml);
            fZl = fZl * __expf(fml - mm2) + oZl * __expf(oml - mm2);
            fml = mm2;
        }
        float lse = fml + logf(fZl);
        row_loss[blockIdx.x * 16 + r] = lse - fW / fZ;   // cross-entropy of row
    }
}

// Deterministic final mean over B row losses.
__global__ void final_reduce(const float* __restrict__ row_loss, int B,
                             float* __restrict__ out) {
    __shared__ float sm[256];
    float acc = 0.0f;
    for (int i = threadIdx.x; i < B; i += 256) acc += row_loss[i];
    sm[threadIdx.x] = acc;
    __syncthreads();
    for (int s = 128; s > 0; s >>= 1) {
        if (threadIdx.x < s) sm[threadIdx.x] += sm[threadIdx.x + s];
        __syncthreads();
    }
    if (threadIdx.x == 0) out[0] = sm[0] / (float)B;
}

extern "C" void kernel_launch(void* const* d_in, const int* in_sizes, int n_in,
                              void* d_out, int out_size, void* d_ws, size_t ws_size,
                              hipStream_t stream) {
    const float* pred   = (const float*)d_in[0];   // [B, C] f32
    const float* latlon = (const float*)d_in[1];   // [B, 2] f32
    const float* cent   = (const float*)d_in[2];   // [C, 2] f32
    // d_in[3] (geocell_indices) is mathematically irrelevant: softmax shift
    // invariance cancels dist_n exactly.

    const int B = in_sizes[1] / 2;   // 4096
    const int C = in_sizes[2] / 2;   // 10000 (multiple of 16)

    float* Uv       = (float*)d_ws;                // B*4 floats
    float* Vv       = Uv + (size_t)B * 4;          // C*4 floats
    float* row_loss = Vv + (size_t)C * 4;          // B floats  (~240 KB total)

    prep_unitvec<<<(B + 255) / 256, 256, 0, stream>>>(latlon, B, Uv);
    prep_unitvec<<<(C + 255) / 256, 256, 0, stream>>>(cent, C, Vv);
    hav_loss_main<<<B / 16, 128, 0, stream>>>(pred, Uv, Vv, row_loss, C);
    final_reduce<<<1, 256, 0, stream>>>(row_loss, B, (float*)d_out);
}